// KernelConv_80668075753604
// MI455X (gfx1250) — compile-verified
//
#include <hip/hip_runtime.h>
#include <math.h>

// Problem constants (from reference setup_inputs)
#define SS 4          // num supports
#define DD 3          // spatial dim
#define FF 32         // feature dim
#define EE 16         // edge feature dim
#define LL 32         // num kernels
#define NPERM 24      // 4! permutations
#define PROWS 32      // perms padded to 32 -> two 16-row WMMA tiles per l
#define KDIM 128      // S*F contraction length
#define TNB 64        // neighbors per block (4 WMMA N-tiles)
#define LDA 132       // LDS row stride in floats (128 data + 4 pad, conflict-free)
#define GX 98         // persistent grid width (blocks stride over neighbor groups)

typedef __attribute__((ext_vector_type(2))) float v2f;
typedef __attribute__((ext_vector_type(8))) float v8f;
typedef __attribute__((ext_vector_type(4))) unsigned int v4u;
typedef __attribute__((ext_vector_type(8))) int v8i;
typedef __attribute__((ext_vector_type(4))) int v4i;

#if __has_builtin(__builtin_amdgcn_tensor_load_to_lds)
#define HAVE_TDM 1
#else
#define HAVE_TDM 0
#endif

// itertools.permutations(range(4)) in lexicographic order
__constant__ int c_perms[NPERM][SS] = {
 {0,1,2,3},{0,1,3,2},{0,2,1,3},{0,2,3,1},{0,3,1,2},{0,3,2,1},
 {1,0,2,3},{1,0,3,2},{1,2,0,3},{1,2,3,0},{1,3,0,2},{1,3,2,0},
 {2,0,1,3},{2,0,3,1},{2,1,0,3},{2,1,3,0},{2,3,0,1},{2,3,1,0},
 {3,0,1,2},{3,0,2,1},{3,1,0,2},{3,1,2,0},{3,2,0,1},{3,2,1,0}};

#if HAVE_TDM
// Issue one TDM DMA: 2D tile (rowlen_dw x rows of 4-byte elems), row-major with
// tensor_dim0_stride == rowlen_dw, padded in LDS by 4 DWORDs every 128 DWORDs
// (pad_interval code 6 -> 128 DW, pad_amount code 3 -> 4 DW) => LDS stride 132.
__device__ __forceinline__ void tdm_load_2d(unsigned lds_off, const void* gptr,
                                            unsigned rows, unsigned rowlen_dw) {
    const unsigned long long ga = (unsigned long long)(uintptr_t)gptr;
    v4u g0;
    g0.x = 1u;                                     // count=1, user descriptor
    g0.y = lds_off;                                // lds_addr (bytes)
    g0.z = (unsigned)(ga & 0xFFFFFFFFu);           // global_addr[31:0]
    g0.w = (unsigned)((ga >> 32) & 0x1FFFFFFu) | (2u << 30);  // addr[56:32] | type=2
    v8i g1;
    // workgroup_mask=0 | data_size=2(4B)<<16 | pad_enable<<20 | pad_interval=6<<22 | pad_amount=3<<25
    g1[0] = (int)((2u << 16) | (1u << 20) | (6u << 22) | (3u << 25));
    g1[1] = (int)((rowlen_dw & 0xFFFFu) << 16);    // barrier_addr=0 | tensor_dim0 lo16
    g1[2] = (int)(((rowlen_dw >> 16) & 0xFFFFu) | ((rows & 0xFFFFu) << 16)); // td0 hi | td1 lo
    g1[3] = (int)(((rows >> 16) & 0xFFFFu) | ((rowlen_dw & 0xFFFFu) << 16)); // td1 hi | tile_dim0
    g1[4] = (int)(rows & 0xFFFFu);                 // tile_dim1 | tile_dim2=0
    g1[5] = (int)rowlen_dw;                        // tensor_dim0_stride[31:0]
    g1[6] = 0;                                     // stride0 hi | tensor_dim1_stride lo
    g1[7] = 0;                                     // tensor_dim1_stride hi
    const v4i z4 = {0, 0, 0, 0};                   // groups 2/3 unused (2D tensor)
    const v8i z8 = {0, 0, 0, 0, 0, 0, 0, 0};       // extra group (6-arg builtin), unused
    __builtin_amdgcn_tensor_load_to_lds(g0, g1, z4, z4, z8, 0);
}
#endif

// Build padded permuted-support matrix A[1024][128] and its row norms^2.
__global__ void prep_support(const float* __restrict__ x_support,
                             float* __restrict__ A,
                             float* __restrict__ sq_sup) {
    const int row = blockIdx.x;        // 0..1023
    const int l  = row >> 5;
    const int pr = row & 31;
    const int p  = (pr < NPERM) ? pr : 0;
    const int c  = threadIdx.x;        // 0..127
    const int s  = c >> 5;
    const int f  = c & 31;
    const float v = x_support[(l * SS + c_perms[p][s]) * FF + f];
    A[row * KDIM + c] = v;
    __shared__ float red[KDIM];
    red[c] = v * v;
    __syncthreads();
    for (int off = KDIM / 2; off > 0; off >>= 1) {
        if (c < off) red[c] += red[c + off];
        __syncthreads();
    }
    if (c == 0) sq_sup[row] = red[0];
}

// Per-neighbor precompute: ||x_neighbor||^2, relative-position lengths,
// and roll-by-1 cosine similarities.
__global__ void prep_neighbor(const float* __restrict__ xn,
                              const float* __restrict__ p_nei,
                              const float* __restrict__ p_foc,
                              float* __restrict__ sq_nei,
                              float* __restrict__ len_nei,
                              float* __restrict__ intra_nei,
                              int N) {
    const int n = blockIdx.x * blockDim.x + threadIdx.x;
    if (n >= N) return;
    float sq = 0.0f;
    for (int k = 0; k < KDIM; ++k) {
        const float v = xn[(size_t)n * KDIM + k];
        sq += v * v;
    }
    sq_nei[n] = sq;
    float pn[SS][DD], nr[SS];
    #pragma unroll
    for (int s = 0; s < SS; ++s) {
        float acc = 0.0f;
        #pragma unroll
        for (int d = 0; d < DD; ++d) {
            const float v = p_nei[((size_t)n * SS + s) * DD + d] - p_foc[(size_t)n * DD + d];
            pn[s][d] = v;
            acc += v * v;
        }
        nr[s] = sqrtf(acc);
        len_nei[(size_t)n * SS + s] = nr[s];
    }
    #pragma unroll
    for (int s = 0; s < SS; ++s) {
        const int q = (s + 3) & 3;   // roll(+1): prev element
        float dot = 0.0f;
        #pragma unroll
        for (int d = 0; d < DD; ++d) dot += pn[q][d] * pn[s][d];
        const float dn = fmaxf(nr[q], 1e-8f) * fmaxf(nr[s], 1e-8f);
        intra_nei[(size_t)n * SS + s] = dot / dn;
    }
}

// Persistent main kernel: block = (chunk, l). 8 wave32s as 2(M) x 4(N) WMMA
// tiles. A tile staged once per block; B tiles (64 neighbors x 128) double-
// buffered via TDM tensor_load_to_lds while WMMA consumes the current buffer.
__global__ void __launch_bounds__(256)
score_kernel(const float* __restrict__ A,
             const float* __restrict__ sq_sup,
             const float* __restrict__ xn,
             const float* __restrict__ sq_nei,
             const float* __restrict__ len_nei,
             const float* __restrict__ intra_nei,
             const float* __restrict__ x_focal,
             const float* __restrict__ x_center,
             const float* __restrict__ edge_nei,
             const float* __restrict__ edge_sup,
             const float* __restrict__ p_support,
             float* __restrict__ out, int N) {
    __shared__ float ldsA[PROWS * LDA];          // 16.9 KB
    __shared__ float ldsB[2][TNB * LDA];         // 67.6 KB, double buffered
    __shared__ float dtile[PROWS * (TNB + 1)];   // 8.3 KB
    __shared__ float ldsSqS[PROWS];
    __shared__ float ldsXC[FF];
    __shared__ float ldsPS[SS * DD];
    __shared__ float ldsES[SS * EE];

    const int l    = blockIdx.y;
    const int tid  = threadIdx.x;      // 0..255
    const int w    = tid >> 5;         // wave id (wave32)
    const int wm   = w & 1;            // WMMA M-tile (perm rows)
    const int wn   = w >> 1;           // WMMA N-tile (neighbors)
    const int lane = tid & 31;
    const int hf   = lane >> 4;        // 0 -> K{0,1}, 1 -> K{2,3}
    const int lr   = lane & 15;

    const unsigned NG = (unsigned)((N + TNB - 1) / TNB);

    // one-time small tables
    if (tid < PROWS)   ldsSqS[tid] = sq_sup[l * PROWS + tid];
    if (tid < FF)      ldsXC[tid]  = x_center[l * FF + tid];
    if (tid < SS * DD) ldsPS[tid]  = p_support[l * SS * DD + tid];
    if (tid < SS * EE) ldsES[tid]  = edge_sup[l * SS * EE + tid];

    unsigned g = blockIdx.x;
#if HAVE_TDM
    if (g < NG && w == 0) {
        // Stage A once and the first B tile via Tensor Data Mover (padded rows).
        tdm_load_2d((unsigned)(uintptr_t)&ldsA[0], A + (size_t)l * PROWS * KDIM,
                    PROWS, KDIM);
        const unsigned rows0 = (N - g * TNB < TNB) ? (unsigned)(N - g * TNB) : TNB;
        tdm_load_2d((unsigned)(uintptr_t)&ldsB[0][0], xn + (size_t)g * TNB * KDIM,
                    rows0, KDIM);
        __builtin_amdgcn_s_wait_tensorcnt(0);
    }
#else
    for (int i = tid; i < PROWS * (KDIM / 4); i += 256) {
        const int r = i >> 5, c4 = i & 31;
        const float4 v = ((const float4*)(A + ((size_t)l * PROWS + r) * KDIM))[c4];
        float* dst = &ldsA[r * LDA + c4 * 4];
        dst[0] = v.x; dst[1] = v.y; dst[2] = v.z; dst[3] = v.w;
    }
    if (g < NG) {
        for (int i = tid; i < TNB * (KDIM / 4); i += 256) {
            const int r = i >> 5, c4 = i & 31;
            int nb = g * TNB + r; if (nb >= N) nb = N - 1;
            const float4 v = ((const float4*)(xn + (size_t)nb * KDIM))[c4];
            float* dst = &ldsB[0][r * LDA + c4 * 4];
            dst[0] = v.x; dst[1] = v.y; dst[2] = v.z; dst[3] = v.w;
        }
    }
#endif
    __syncthreads();

    int buf = 0;
    while (g < NG) {
        const unsigned gn = g + GX;
#if HAVE_TDM
        if (gn < NG && w == 0) {   // prefetch next B tile while computing
            const unsigned rows = (N - gn * TNB < TNB) ? (unsigned)(N - gn * TNB) : TNB;
            tdm_load_2d((unsigned)(uintptr_t)&ldsB[buf ^ 1][0],
                        xn + (size_t)gn * TNB * KDIM, rows, KDIM);
        }
#endif
        const int n0 = (int)g * TNB;

        // --- f32 WMMA GEMM: cross[m][n], K=128, per-wave 16x16 tile ---
        // A(16x4) frag: lanes 0-15 (M=lane, K=0/1), lanes 16-31 (M=lane-16, K=2/3)
        // B(4x16) frag: lanes 0-15 (N=lane, K=0/1), lanes 16-31 (N=lane-16, K=2/3)
        v8f acc = {};
        const float* Bb = &ldsB[buf][0];
        const int abase = (wm * 16 + lr) * LDA;
        const int bbase = (wn * 16 + lr) * LDA;
        #pragma unroll
        for (int k = 0; k < KDIM / 4; ++k) {
            const int koff = 4 * k + 2 * hf;
            const v2f a = *(const v2f*)&ldsA[abase + koff];
            const v2f b = *(const v2f*)&Bb[bbase + koff];
            acc = __builtin_amdgcn_wmma_f32_16x16x4_f32(
                false, a, false, b, (short)0, acc, false, false);
        }

        // d = ||sup||^2 + ||nei||^2 - 2*cross, clamped at 0
        int nc = n0 + wn * 16 + lr; if (nc >= N) nc = N - 1;
        const float sqn = sq_nei[nc];
        #pragma unroll
        for (int v = 0; v < 8; ++v) {
            const int row = wm * 16 + hf * 8 + v;
            const float dv = fmaxf(ldsSqS[row] + sqn - 2.0f * acc[v], 0.0f);
            dtile[row * (TNB + 1) + wn * 16 + lr] = dv;
        }
        __syncthreads();

        // --- epilogue: one thread per neighbor column ---
        if (tid < TNB) {
            const int n = n0 + tid;
            if (n < N) {
                // argmin of d over perms == argmax of atan(1/d) (monotone)
                float dmin = 3.402823466e38f; int br = 0;
                for (int r = 0; r < PROWS; ++r) {
                    const float v = dtile[r * (TNB + 1) + tid];
                    if (v < dmin) { dmin = v; br = r; }
                }
                const int p = (br < NPERM) ? br : 0;
                const float HP = 1.57079632679489662f;
                const float sc_attr = atanf(1.0f / dmin);   // d=0 -> +inf -> pi/2

                float bp[SS][DD], ln[SS];
                #pragma unroll
                for (int s = 0; s < SS; ++s) {
                    const int idx = c_perms[p][s];
                    float a2 = 0.0f;
                    #pragma unroll
                    for (int d = 0; d < DD; ++d) {
                        const float v = ldsPS[idx * DD + d];
                        bp[s][d] = v; a2 += v * v;
                    }
                    ln[s] = sqrtf(a2);
                }
                float asum = 0.0f, lsum = 0.0f;
                #pragma unroll
                for (int s = 0; s < SS; ++s) {
                    const int q = (s + 3) & 3;
                    float dot = 0.0f;
                    #pragma unroll
                    for (int d = 0; d < DD; ++d) dot += bp[q][d] * bp[s][d];
                    const float cs = dot / (fmaxf(ln[q], 1e-8f) * fmaxf(ln[s], 1e-8f));
                    const float da = intra_nei[(size_t)n * SS + s] - cs;
                    asum += da * da;
                    const float dl = len_nei[(size_t)n * SS + s] - ln[s];
                    lsum += dl * dl;
                }
                const float sc_ang = atanf(1.0f / asum);
                const float sc_len = atanf(1.0f / lsum);

                float csum = 0.0f;
                for (int f = 0; f < FF; ++f) {
                    const float d0 = x_focal[(size_t)n * FF + f] - ldsXC[f];
                    csum += d0 * d0;
                }
                const float sc_cen = atanf(1.0f / csum);

                float esum = 0.0f;
                #pragma unroll
                for (int s = 0; s < SS; ++s) {
                    const int idx = c_perms[p][s];
                    for (int e = 0; e < EE; ++e) {
                        const float d0 = edge_nei[((size_t)n * SS + s) * EE + e]
                                       - ldsES[idx * EE + e];
                        esum += d0 * d0;
                    }
                }
                const float sc_edge = atanf(1.0f / esum);

                float t, tot = 0.0f;
                t = sc_len  - HP; tot += t * t;
                t = sc_ang  - HP; tot += t * t;
                t = sc_attr - HP; tot += t * t;
                t = sc_cen  - HP; tot += t * t;
                t = sc_edge - HP; tot += t * t;
                out[(size_t)l * N + n] = atanf(1.0f / tot);
            }
        }

#if HAVE_TDM
        if (w == 0) __builtin_amdgcn_s_wait_tensorcnt(0);  // next buffer landed
#else
        if (gn < NG) {
            __syncthreads();
            for (int i = tid; i < TNB * (KDIM / 4); i += 256) {
                const int r = i >> 5, c4 = i & 31;
                int nb = gn * TNB + r; if (nb >= N) nb = N - 1;
                const float4 v = ((const float4*)(xn + (size_t)nb * KDIM))[c4];
                float* dst = &ldsB[buf ^ 1][r * LDA + c4 * 4];
                dst[0] = v.x; dst[1] = v.y; dst[2] = v.z; dst[3] = v.w;
            }
        }
#endif
        __syncthreads();   // dtile consumed + next buffer visible to all waves
        buf ^= 1;
        g = gn;
    }
}

extern "C" void kernel_launch(void* const* d_in, const int* in_sizes, int n_in,
                              void* d_out, int out_size, void* d_ws, size_t ws_size,
                              hipStream_t stream) {
    (void)n_in; (void)out_size; (void)ws_size;
    const float* x_focal  = (const float*)d_in[0];  // [N,32]
    const float* p_focal  = (const float*)d_in[1];  // [N,3]
    const float* x_nei    = (const float*)d_in[2];  // [N,4,32]
    const float* p_nei    = (const float*)d_in[3];  // [N,4,3]
    const float* e_nei    = (const float*)d_in[4];  // [N,4,16]
    const float* x_center = (const float*)d_in[5];  // [32,1,32]
    const float* x_sup    = (const float*)d_in[6];  // [32,4,32]
    const float* e_sup    = (const float*)d_in[7];  // [32,4,16]
    const float* p_sup    = (const float*)d_in[8];  // [32,4,3]
    float* out = (float*)d_out;                     // [32,N]
    const int N = in_sizes[0] / FF;

    // workspace layout (floats)
    float* ws      = (float*)d_ws;
    float* A       = ws;                          // 1024*128
    float* sq_sup  = A + LL * PROWS * KDIM;       // 1024
    float* sq_nei  = sq_sup + LL * PROWS;         // N
    float* len_nei = sq_nei + N;                  // N*4
    float* intra   = len_nei + (size_t)N * SS;    // N*4

    prep_support<<<dim3(LL * PROWS), dim3(KDIM), 0, stream>>>(x_sup, A, sq_sup);
    prep_neighbor<<<dim3((N + 255) / 256), dim3(256), 0, stream>>>(
        x_nei, p_nei, p_focal, sq_nei, len_nei, intra, N);

    dim3 grid(GX, LL);
    score_kernel<<<grid, dim3(256), 0, stream>>>(
        A, sq_sup, x_nei, sq_nei, len_nei, intra,
        x_focal, x_center, e_nei, e_sup, p_sup, out, N);
}